// _ECELoss_48567490183751
// MI455X (gfx1250) — compile-verified
//
#include <hip/hip_runtime.h>
#include <hip/hip_bf16.h>
#include <cstdint>

#define NBINS 15
#define BLOCK 256
#define WAVES (BLOCK / 32)
#define MAX_GRID 4096
#define CHUNK 128                 // elements per wave-chunk (4 per lane, wave32)
#define CHUNK_BYTES (CHUNK * 4)   // 512 B per stream per chunk

// clang ext-vectors (HIP's int4/float4 are structs; these lower to b128 ops)
typedef int   v4i __attribute__((ext_vector_type(4)));
typedef float v4f __attribute__((ext_vector_type(4)));

// ---------------------------------------------------------------------------
// CDNA5 async global->LDS copy (tracked by ASYNCcnt), GV addressing mode.
// vdst = VGPR holding workgroup-relative LDS byte address
// vaddr = 64-bit global address in a VGPR pair
// ---------------------------------------------------------------------------
__device__ __forceinline__ void async_load_b128(unsigned lds_off,
                                                unsigned long long gaddr) {
    asm volatile("global_load_async_to_lds_b128 %0, %1, off th:TH_LOAD_NT"
                 :: "v"(lds_off), "v"(gaddr)
                 : "memory");
}
__device__ __forceinline__ void wait_async_le(int) = delete;
__device__ __forceinline__ void wait_async_le3() {
    asm volatile("s_wait_asynccnt 0x3" ::: "memory");
}
__device__ __forceinline__ void wait_async_0() {
    asm volatile("s_wait_asynccnt 0x0" ::: "memory");
}

// ---------------------------------------------------------------------------
// Workspace layout (d_ws):
//   [0,   128): double g_conf[15]   (conf-sum per bin, f64 merge)
//   [128, 192): uint   g_cnt[15]    (count per bin, exact)
//   [192, 256): uint   g_acc[15]    (correct-count per bin, exact)
// ---------------------------------------------------------------------------

__global__ void ece_zero(double* __restrict__ g_conf,
                         unsigned* __restrict__ g_cnt,
                         unsigned* __restrict__ g_acc) {
    int i = threadIdx.x;
    if (i < NBINS) {
        g_conf[i] = 0.0;
        g_cnt[i]  = 0u;
        g_acc[i]  = 0u;
    }
}

__global__ __launch_bounds__(BLOCK) void ece_hist(
    const int* __restrict__ preds, const int* __restrict__ targets,
    const float* __restrict__ confs, int n,
    double* __restrict__ g_conf, unsigned* __restrict__ g_cnt,
    unsigned* __restrict__ g_acc)
{
    // double-buffered async staging: [wave][buf][stream: preds,targets,confs]
    __shared__ __align__(16) char stage[WAVES][2][3][CHUNK_BYTES];
    // per-wave private histograms: no inter-wave ds contention (wave32)
    __shared__ float    s_conf[WAVES][NBINS];
    __shared__ unsigned s_cnt [WAVES][NBINS];
    __shared__ unsigned s_acc [WAVES][NBINS];

    const int wave = threadIdx.x >> 5;
    const int lane = threadIdx.x & 31;

    for (int i = threadIdx.x; i < WAVES * NBINS; i += BLOCK) {
        (&s_conf[0][0])[i] = 0.0f;
        (&s_cnt [0][0])[i] = 0u;
        (&s_acc [0][0])[i] = 0u;
    }
    __syncthreads();

    const int gwave   = blockIdx.x * WAVES + wave;
    const int wstride = gridDim.x * WAVES;
    const int nchunks = n / CHUNK;

    const unsigned long long pb = (unsigned long long)(uintptr_t)preds;
    const unsigned long long tb = (unsigned long long)(uintptr_t)targets;
    const unsigned long long cb = (unsigned long long)(uintptr_t)confs;

    // Workgroup-relative LDS byte addresses for this lane's 16B slice.
    // (flat LDS pointer: aperture in [63:32], LDS offset in [31:0])
    unsigned lds_base[2][3];
#pragma unroll
    for (int b = 0; b < 2; ++b)
#pragma unroll
        for (int s = 0; s < 3; ++s)
            lds_base[b][s] =
                (unsigned)(uintptr_t)(&stage[wave][b][s][0]) + (unsigned)(lane * 16);

    auto issue = [&](int buf, int chunk) {
        unsigned long long off =
            (unsigned long long)chunk * CHUNK_BYTES + (unsigned)(lane * 16);
        async_load_b128(lds_base[buf][0], pb + off);
        async_load_b128(lds_base[buf][1], tb + off);
        async_load_b128(lds_base[buf][2], cb + off);
    };

    int c   = gwave;
    int cur = 0;
    if (c < nchunks) issue(cur, c);           // prologue: fill buffer 0
    while (c < nchunks) {
        const int nxt = c + wstride;
        if (nxt < nchunks) {
            issue(cur ^ 1, nxt);              // keep next chunk in flight
            wait_async_le3();                 // in-order: current chunk landed
        } else {
            wait_async_0();
        }
        const v4i p  = *(const v4i*)(&stage[wave][cur][0][lane * 16]);
        const v4i t  = *(const v4i*)(&stage[wave][cur][1][lane * 16]);
        const v4f cf = *(const v4f*)(&stage[wave][cur][2][lane * 16]);
#pragma unroll
        for (int j = 0; j < 4; ++j) {
            float cfv = cf[j];
            // bin i covers (i/15, (i+1)/15]; conf <= 0 -> no bin (idx = -1)
            int idx = (int)ceilf(cfv * 15.0f) - 1;
            if (idx >= 0) {
                atomicAdd(&s_cnt [wave][idx], 1u);
                atomicAdd(&s_conf[wave][idx], cfv);
                if (p[j] == t[j]) atomicAdd(&s_acc[wave][idx], 1u);
            }
        }
        c = nxt;
        cur ^= 1;
    }

    // scalar tail over [nchunks*CHUNK, n)
    const int tid     = blockIdx.x * BLOCK + threadIdx.x;
    const int tstride = gridDim.x * BLOCK;
    for (int i = nchunks * CHUNK + tid; i < n; i += tstride) {
        float cfv = confs[i];
        int idx = (int)ceilf(cfv * 15.0f) - 1;
        if (idx >= 0) {
            atomicAdd(&s_cnt [wave][idx], 1u);
            atomicAdd(&s_conf[wave][idx], cfv);
            if (preds[i] == targets[i]) atomicAdd(&s_acc[wave][idx], 1u);
        }
    }
    __syncthreads();

    // merge the 8 wave-local copies, one global atomic per bin per block
    for (int b = threadIdx.x; b < NBINS; b += BLOCK) {
        double   cs = 0.0;
        unsigned cn = 0, ac = 0;
        for (int w = 0; w < WAVES; ++w) {
            cs += (double)s_conf[w][b];
            cn += s_cnt[w][b];
            ac += s_acc[w][b];
        }
        atomicAdd(&g_conf[b], cs);   // global_atomic_add_f64
        atomicAdd(&g_cnt [b], cn);   // global_atomic_add_u32
        atomicAdd(&g_acc [b], ac);
    }
}

__global__ void ece_final(const double* __restrict__ g_conf,
                          const unsigned* __restrict__ g_cnt,
                          const unsigned* __restrict__ g_acc,
                          float* __restrict__ out, int n)
{
    if (threadIdx.x == 0) {
        float ece = 0.0f;
        float fn  = (float)n;
        for (int b = 0; b < NBINS; ++b) {
            float cnt = (float)g_cnt[b];
            if (cnt > 0.0f) {
                float conf_mean = (float)g_conf[b] / cnt;
                float acc_mean  = (float)g_acc[b]  / cnt;
                float gap  = fabsf(conf_mean - acc_mean);
                ece += gap * (cnt / fn);   // gap * prop over non-empty bins
            }
        }
        out[0] = ece;
    }
}

extern "C" void kernel_launch(void* const* d_in, const int* in_sizes, int n_in,
                              void* d_out, int out_size, void* d_ws, size_t ws_size,
                              hipStream_t stream) {
    const int*   preds   = (const int*)d_in[0];
    const int*   targets = (const int*)d_in[1];
    const float* confs   = (const float*)d_in[2];
    const int    n       = in_sizes[2];

    char*     ws     = (char*)d_ws;
    double*   g_conf = (double*)(ws);
    unsigned* g_cnt  = (unsigned*)(ws + 128);
    unsigned* g_acc  = (unsigned*)(ws + 192);

    ece_zero<<<1, 32, 0, stream>>>(g_conf, g_cnt, g_acc);

    int nchunks = n / CHUNK;
    int grid    = (nchunks + WAVES - 1) / WAVES;
    if (grid > MAX_GRID) grid = MAX_GRID;
    if (grid < 1)        grid = 1;

    ece_hist<<<grid, BLOCK, 0, stream>>>(preds, targets, confs, n,
                                         g_conf, g_cnt, g_acc);

    ece_final<<<1, 32, 0, stream>>>(g_conf, g_cnt, g_acc, (float*)d_out, n);
}